// LearnableGaugeCovariantBlocker_42004780155662
// MI455X (gfx1250) — compile-verified
//
#include <hip/hip_runtime.h>
#include <stdint.h>

// Problem constants (fixed by the reference: B=8, L=2048, coarse 1024x1024).
#define L_FINE   2048
#define L_MASK   2047
#define L_SHIFT  11
#define L_COARSE 1024
#define CH 8                 // coarse rows per block
#define CW 32                // coarse cols per block
#define TR (2*CH + 2)        // 18 fine rows in tile   (window rows -1..+2*CH)
#define TC (2*CW + 2)        // 66 fine cols in tile
#define TCP 68               // padded LDS row stride (floats)
#define NLOAD (TR * TC)      // 1188 elements per field

// Floored modulo (jnp.mod semantics): mod(a,n) = a - floor(a/n)*n, n = 2*pi.
// Valid here: |path| <= ~sum of 4 gaussians, tiny vs fp32 range, so the
// divide-free floor form is exact to fp32 rounding. 4 VALU ops vs fmodf's
// ~100-op general range reduction.
__device__ __forceinline__ float regularize_f(float x) {
    const float PI_F      = 3.14159265358979323846f;
    const float TWO_PI_F  = 6.28318530717958647692f;
    const float INV_2PI_F = 0.15915494309189533577f;
    float a = x + PI_F;
    float m = a - floorf(a * INV_2PI_F) * TWO_PI_F;   // in [0, 2*pi)
    return m - PI_F;
}

__global__ __launch_bounds__(256)
void gauge_blocker_kernel(const float* __restrict__ fine,
                          const float* __restrict__ logits,
                          float* __restrict__ out)
{
    __shared__ float sux[TR * TCP];
    __shared__ float suy[TR * TCP];

    const int tx  = threadIdx.x;          // 0..31
    const int ty  = threadIdx.y;          // 0..7
    const int tid = ty * 32 + tx;         // 0..255
    const int j0  = blockIdx.x * CW;      // coarse col origin
    const int i0  = blockIdx.y * CH;      // coarse row origin
    const int b   = blockIdx.z;           // batch

    const int r0 = 2 * i0 - 1;            // fine-tile origin (row)
    const int c0 = 2 * j0 - 1;            // fine-tile origin (col)

    const uint64_t planeUx = (uint64_t)(uintptr_t)
        (fine + ((size_t)b * 2 + 0) * (size_t)L_FINE * L_FINE);
    const uint64_t planeUy = planeUx + (uint64_t)L_FINE * L_FINE * sizeof(float);

    // LDS byte offsets of the shared tiles (low 32 bits of the flat LDS
    // address are the wave-relative LDS offset on gfx1250).
    const uint32_t ldsUx = (uint32_t)(uintptr_t)(void*)sux;
    const uint32_t ldsUy = (uint32_t)(uintptr_t)(void*)suy;

    // ---- Issue async global -> LDS tile loads (CDNA5 async data mover) ----
    for (int k = tid; k < NLOAD; k += 256) {
        int lr = k / TC;
        int lc = k - lr * TC;
        uint32_t rr = (uint32_t)(r0 + lr + L_FINE) & L_MASK;   // torus wrap
        uint32_t cc = (uint32_t)(c0 + lc + L_FINE) & L_MASK;
        uint32_t goff = ((rr << L_SHIFT) + cc) << 2;           // byte offset in plane
        uint32_t dUx  = ldsUx + (uint32_t)(lr * TCP + lc) * 4u;
        uint32_t dUy  = ldsUy + (uint32_t)(lr * TCP + lc) * 4u;
        asm volatile("global_load_async_to_lds_b32 %0, %1, %2"
                     :: "v"(dUx), "v"(goff), "s"(planeUx) : "memory");
        asm volatile("global_load_async_to_lds_b32 %0, %1, %2"
                     :: "v"(dUy), "v"(goff), "s"(planeUy) : "memory");
    }

    // ---- Softmax path weights: uniform scalar work, overlapped with the
    //      in-flight async loads (before s_wait_asynccnt). ----
    float wx0, wx1, wx2, wy0, wy1, wy2;
    {
        float a0 = logits[0], a1 = logits[1], a2 = logits[2];
        float m  = fmaxf(a0, fmaxf(a1, a2));
        float e0 = __expf(a0 - m), e1 = __expf(a1 - m), e2 = __expf(a2 - m);
        float inv = 1.0f / (e0 + e1 + e2);
        wx0 = e0 * inv; wx1 = e1 * inv; wx2 = e2 * inv;
        a0 = logits[3]; a1 = logits[4]; a2 = logits[5];
        m  = fmaxf(a0, fmaxf(a1, a2));
        e0 = __expf(a0 - m); e1 = __expf(a1 - m); e2 = __expf(a2 - m);
        inv = 1.0f / (e0 + e1 + e2);
        wy0 = e0 * inv; wy1 = e1 * inv; wy2 = e2 * inv;
    }

    asm volatile("s_wait_asynccnt 0x0" ::: "memory");
    __syncthreads();

    // ---- Stencil from LDS (local window around fine site (2i,2j)) ----
    const int lr = 2 * ty + 1;   // tile-local row of fine center (origin at -1)
    const int lc = 2 * tx + 1;
    const float* ux = &sux[lr * TCP + lc];
    const float* uy = &suy[lr * TCP + lc];
#define AT(p, dr, dc) (p)[(dr) * TCP + (dc)]

    // x-paths: straight, up, down   (roll conventions verified against jnp.roll)
    float pxs =  AT(ux,0,0) + AT(ux,1,0);
    float pxu =  AT(uy,0,0) + AT(ux,0,1)  + AT(ux,1,1)  - AT(uy,2,0);
    float pxd = -AT(uy,0,-1) + AT(ux,0,-1) + AT(ux,1,-1) + AT(uy,2,-1);
    // y-paths: straight, right, left
    float pys =  AT(uy,0,0) + AT(uy,0,1);
    float pyr =  AT(ux,0,0) + AT(uy,1,0)  + AT(uy,1,1)  - AT(ux,0,2);
    float pyl = -AT(ux,-1,0) + AT(uy,-1,0) + AT(uy,-1,1) + AT(ux,-1,2);
#undef AT

    // ---- Circular average ----
    float s, c, ss, cs;
    __sincosf(regularize_f(pxs), &s, &c); ss  = wx0 * s; cs  = wx0 * c;
    __sincosf(regularize_f(pxu), &s, &c); ss += wx1 * s; cs += wx1 * c;
    __sincosf(regularize_f(pxd), &s, &c); ss += wx2 * s; cs += wx2 * c;
    float ox = regularize_f(atan2f(ss, cs));

    __sincosf(regularize_f(pys), &s, &c); ss  = wy0 * s; cs  = wy0 * c;
    __sincosf(regularize_f(pyr), &s, &c); ss += wy1 * s; cs += wy1 * c;
    __sincosf(regularize_f(pyl), &s, &c); ss += wy2 * s; cs += wy2 * c;
    float oy = regularize_f(atan2f(ss, cs));

    // ---- Coalesced non-temporal stores (write-once data; keep L2 for the
    //      input stream + inter-block halo reuse). ----
    const int i = i0 + ty, j = j0 + tx;
    size_t ob = ((size_t)b * 2) * (size_t)L_COARSE * L_COARSE
              + (size_t)i * L_COARSE + (size_t)j;
    __builtin_nontemporal_store(ox, &out[ob]);
    __builtin_nontemporal_store(oy, &out[ob + (size_t)L_COARSE * L_COARSE]);
}

extern "C" void kernel_launch(void* const* d_in, const int* in_sizes, int n_in,
                              void* d_out, int out_size, void* d_ws, size_t ws_size,
                              hipStream_t stream) {
    (void)in_sizes; (void)n_in; (void)out_size; (void)d_ws; (void)ws_size;
    const float* fine   = (const float*)d_in[0];   // (8, 2, 2048, 2048) fp32
    const float* logits = (const float*)d_in[1];   // (2, 3) fp32
    float* out          = (float*)d_out;           // (8, 2, 1024, 1024) fp32

    dim3 block(32, 8, 1);
    dim3 grid(L_COARSE / CW, L_COARSE / CH, 8);    // (32, 128, 8)
    gauge_blocker_kernel<<<grid, block, 0, stream>>>(fine, logits, out);
}